// HIERALoss_47742856462442
// MI455X (gfx1250) — compile-verified
//
#include <hip/hip_runtime.h>
#include <cstdint>

// ---------------------------------------------------------------------------
// HIERA loss on MI455X (gfx1250).
// Bandwidth-bound streaming reduction: ~109 MB in -> 1 float out (~4.7 us at
// 23.3 TB/s HBM). No matrix math -> no WMMA; instead we use the CDNA5 Tensor
// Data Mover: one TENSOR_LOAD_TO_LDS per 26x256 tile (26 channel rows, row
// stride = 1 MiB plane), double-buffered in LDS, synchronized with
// s_wait_tensorcnt + workgroup barriers. Fallback: per-lane
// global_load_async_to_lds_b32 (ASYNCcnt).
// ---------------------------------------------------------------------------

#define NCLS   26
#define NPAR   6
#define NLEAF  20
#define PLANE  (512 * 512)
#define NPIX   (4 * PLANE)
#define CHUNK  256
#define NCHUNKS (NPIX / CHUNK)   // 4096 exactly; 1024 chunks per batch image
#define EPSF   1e-8f
#define MAIN_BLOCKS 1024

typedef __attribute__((address_space(1))) int gas_int;
typedef __attribute__((address_space(3))) int las_int;
typedef unsigned int u32x4 __attribute__((ext_vector_type(4)));
typedef int          i32x4 __attribute__((ext_vector_type(4)));
typedef int          i32x8 __attribute__((ext_vector_type(8)));

// ---- fallback: per-lane async global->LDS copy (ASYNCcnt) -----------------
__device__ __forceinline__ void async_copy_b32(const float* gsrc, float* ldst) {
#if __has_builtin(__builtin_amdgcn_global_load_async_to_lds_b32)
  __builtin_amdgcn_global_load_async_to_lds_b32(
      (gas_int*)gsrc, (las_int*)ldst, /*imm offset*/ 0, /*cpol*/ 0);
#else
  unsigned lds_off = (unsigned)(size_t)(void*)ldst;
  asm volatile("global_load_async_to_lds_b32 %0, %1, off"
               :: "v"(lds_off), "v"(gsrc) : "memory");
#endif
}

// ---- TDM: one descriptor moves the whole 26x256 fp32 tile -----------------
#if __has_builtin(__builtin_amdgcn_tensor_load_to_lds)
#define HIERA_USE_TDM 1
__device__ __forceinline__ void tdm_load_tile(const float* gsrc, float* lbase) {
  const uint64_t ga  = (uint64_t)(size_t)gsrc;           // 57-bit byte address
  const unsigned lds = (unsigned)(size_t)(void*)lbase;   // LDS byte offset

  u32x4 g0;
  g0.x = 1u;                                   // count=1, is_restore=0, no gather
  g0.y = lds;                                  // lds_addr
  g0.z = (unsigned)(ga & 0xFFFFFFFFu);         // global_addr[31:0]
  g0.w = (unsigned)((ga >> 32) & 0x01FFFFFFu)  // global_addr[56:32]
         | (2u << 30);                         // type=2 ("image")

  i32x8 g1;
  g1[0] = (int)(2u << 16);                     // wg_mask=0, data_size=2 (4 B)
  g1[1] = (int)((((unsigned)PLANE & 0xFFFFu)) << 16);          // tensor_dim0[15:0]
  g1[2] = (int)((((unsigned)PLANE >> 16) & 0xFFFFu)            // tensor_dim0[31:16]
         | ((unsigned)NCLS << 16));                            // tensor_dim1[15:0]
  g1[3] = (int)((unsigned)CHUNK << 16);        // tensor_dim1[31:16]=0 | tile_dim0=256
  g1[4] = (int)(unsigned)NCLS;                 // tile_dim1=26 | tile_dim2=0
  g1[5] = (int)(unsigned)PLANE;                // tensor_dim0_stride[31:0]
  g1[6] = 0;                                   // stride0[47:32]=0 | stride1[15:0]=0
  g1[7] = 0;                                   // stride1[47:16]=0
  const i32x4 z4 = {0, 0, 0, 0};               // groups 2/3 unused (2D tensor)
  const i32x8 z8 = {0, 0, 0, 0, 0, 0, 0, 0};
  __builtin_amdgcn_tensor_load_to_lds(g0, g1, z4, z4, z8, /*cpol*/ 0);
}
#endif

// ---- tile issue / wait wrappers (wave 0 drives the TDM) -------------------
__device__ __forceinline__ void issue_tile(const float* gp, float* lbase, int tid) {
#if defined(HIERA_USE_TDM)
  if (tid < 32) tdm_load_tile(gp, lbase);      // one wave issues the DMA
#else
#pragma unroll
  for (int c = 0; c < NCLS; ++c)
    async_copy_b32(gp + (size_t)c * PLANE + tid, lbase + c * CHUNK + tid);
#endif
}

__device__ __forceinline__ void wait_tile_pending1(int tid) {
#if defined(HIERA_USE_TDM)
  if (tid < 32) __builtin_amdgcn_s_wait_tensorcnt(1);
#else
#if __has_builtin(__builtin_amdgcn_s_wait_asynccnt)
  __builtin_amdgcn_s_wait_asynccnt(26);
#else
  asm volatile("s_wait_asynccnt 26" ::: "memory");
#endif
#endif
}

__device__ __forceinline__ void wait_tile_0(int tid) {
#if defined(HIERA_USE_TDM)
  if (tid < 32) __builtin_amdgcn_s_wait_tensorcnt(0);
#else
#if __has_builtin(__builtin_amdgcn_s_wait_asynccnt)
  __builtin_amdgcn_s_wait_asynccnt(0);
#else
  asm volatile("s_wait_asynccnt 0" ::: "memory");
#endif
#endif
}

// chunk -> global pointer of its tile origin (chunks never straddle a batch)
__device__ __forceinline__ const float* chunk_base(const float* logits, int chunk) {
  const int pix0 = chunk * CHUNK;
  const int b    = pix0 / PLANE;
  const int hw   = pix0 - b * PLANE;
  return logits + (size_t)b * NCLS * PLANE + hw;
}

// ---------------------------------------------------------------------------
__global__ __launch_bounds__(256) void hiera_main(
    const float* __restrict__ logits, const int* __restrict__ labels,
    double* __restrict__ acc)
{
  __shared__ float tile[2][NCLS * CHUNK];   // 2 x 26 KiB double buffer
  const int tid = threadIdx.x;

  float a_bl = 0.f, a_bp = 0.f, a_ce = 0.f, a_nv = 0.f;

  int chunk = blockIdx.x;
  if (chunk < NCHUNKS)
    issue_tile(chunk_base(logits, chunk), &tile[0][0], tid);

  int buf = 0;
  for (; chunk < NCHUNKS; chunk += gridDim.x) {
    const int nxt = chunk + gridDim.x;
    if (nxt < NCHUNKS) {
      issue_tile(chunk_base(logits, nxt), &tile[buf ^ 1][0], tid);
      wait_tile_pending1(tid);   // current tile has landed, next in flight
    } else {
      wait_tile_0(tid);
    }
    __syncthreads();             // publish wave-0's TDM tile to all 8 waves

    // ---- sigmoid over 26 channels -------------------------------------
    float p[NCLS];
#pragma unroll
    for (int c = 0; c < NCLS; ++c) {
      const float x = tile[buf][c * CHUNK + tid];
      p[c] = 1.0f / (1.0f + __expf(-x));
    }

    const int  lab   = labels[chunk * CHUNK + tid];
    const bool valid = (lab != NCLS);

    // ---- per-parent child max (leaf groups 3,3,3,3,3,5) ---------------
    float cmx[NPAR];
    cmx[0] = fmaxf(fmaxf(p[6],  p[7]),  p[8]);
    cmx[1] = fmaxf(fmaxf(p[9],  p[10]), p[11]);
    cmx[2] = fmaxf(fmaxf(p[12], p[13]), p[14]);
    cmx[3] = fmaxf(fmaxf(p[15], p[16]), p[17]);
    cmx[4] = fmaxf(fmaxf(p[18], p[19]), p[20]);
    cmx[5] = fmaxf(fmaxf(fmaxf(p[21], p[22]), fmaxf(p[23], p[24])), p[25]);

    float mcmaP[NPAR], mclaP[NPAR];
#pragma unroll
    for (int q = 0; q < NPAR; ++q) {
      mclaP[q] = p[q];
      mcmaP[q] = fmaxf(p[q], cmx[q]);
    }

    float mcmaL[NLEAF], mclaL[NLEAF];
#pragma unroll
    for (int l = 0; l < NLEAF; ++l) {
      const int q = (l < 15) ? (l / 3) : 5;    // LEAF_PARENT in closed form
      mcmaL[l] = p[6 + l];
      mclaL[l] = fminf(p[6 + l], p[q]);
    }

    // ---- BCE: -sum log(1-mcma+eps) plus correction at the label class --
    float s1l = 0.f, s1p = 0.f;
#pragma unroll
    for (int l = 0; l < NLEAF; ++l) s1l += __logf(1.f - mcmaL[l] + EPSF);
#pragma unroll
    for (int q = 0; q < NPAR; ++q)  s1p += __logf(1.f - mcmaP[q] + EPSF);

    float selMl = 0.f, selLl = 1.f;            // label-class leaf values
#pragma unroll
    for (int l = 0; l < NLEAF; ++l) {
      const bool m = (lab == 6 + l);
      selMl = m ? mcmaL[l] : selMl;
      selLl = m ? mclaL[l] : selLl;
    }
    const int lpq = (lab < NPAR) ? lab : (((lab - 6) < 15) ? ((lab - 6) / 3) : 5);
    float selMp = 0.f, selLp = 1.f;            // label-class parent values
#pragma unroll
    for (int q = 0; q < NPAR; ++q) {
      const bool m = (lpq == q);
      selMp = m ? mcmaP[q] : selMp;
      selLp = m ? mclaP[q] : selLp;
    }

    const bool isLeaf = valid && (lab >= NPAR);
    float bl = -s1l, bp = -s1p;
    if (isLeaf) bl += __logf(1.f - selMl + EPSF) - __logf(selLl + EPSF);
    if (valid)  bp += __logf(1.f - selMp + EPSF) - __logf(selLp + EPSF);

    // ---- CE: log-softmax over each group, gathered at the label class --
    float mxl = mcmaL[0];
#pragma unroll
    for (int l = 1; l < NLEAF; ++l) mxl = fmaxf(mxl, mcmaL[l]);
    float sse = 0.f;
#pragma unroll
    for (int l = 0; l < NLEAF; ++l) sse += __expf(mcmaL[l] - mxl);
    const float lsl = __logf(sse);

    float mxp = mcmaP[0];
#pragma unroll
    for (int q = 1; q < NPAR; ++q) mxp = fmaxf(mxp, mcmaP[q]);
    float ssp = 0.f;
#pragma unroll
    for (int q = 0; q < NPAR; ++q) ssp += __expf(mcmaP[q] - mxp);
    const float lsp = __logf(ssp);

    float ce = 0.f;
    if (isLeaf) ce += -(selMl - mxl - lsl);
    if (valid)  ce += -(selMp - mxp - lsp);

    if (valid) { a_bl += bl; a_bp += bp; a_nv += 1.f; }
    a_ce += ce;

    __syncthreads();             // all reads of tile[buf] done before reuse
    buf ^= 1;
  }

  // ---- wave32 shuffle reduce, then four f64 atomics per block ------------
#pragma unroll
  for (int off = 16; off > 0; off >>= 1) {
    a_bl += __shfl_down(a_bl, off);
    a_bp += __shfl_down(a_bp, off);
    a_ce += __shfl_down(a_ce, off);
    a_nv += __shfl_down(a_nv, off);
  }
  __shared__ float red[4][8];
  const int wave = tid >> 5;
  if ((tid & 31) == 0) {
    red[0][wave] = a_bl; red[1][wave] = a_bp;
    red[2][wave] = a_ce; red[3][wave] = a_nv;
  }
  __syncthreads();
  if (tid == 0) {
    float bl = 0.f, bp = 0.f, ce = 0.f, nv = 0.f;
#pragma unroll
    for (int w = 0; w < 8; ++w) {
      bl += red[0][w]; bp += red[1][w]; ce += red[2][w]; nv += red[3][w];
    }
    atomicAdd(&acc[0], (double)bl);
    atomicAdd(&acc[1], (double)bp);
    atomicAdd(&acc[2], (double)ce);
    atomicAdd(&acc[3], (double)nv);
  }
}

__global__ void hiera_init(double* __restrict__ acc) {
  if (threadIdx.x < 4) acc[threadIdx.x] = 0.0;
}

__global__ void hiera_final(const double* __restrict__ acc,
                            float* __restrict__ out) {
  if (threadIdx.x == 0) {
    const double bl = acc[0], bp = acc[1], ce = acc[2], nv = acc[3];
    const double loss = bl / nv / 20.0 + bp / nv / 6.0;
    out[0] = (float)(5.0 * loss + ce / (double)NPIX);
  }
}

extern "C" void kernel_launch(void* const* d_in, const int* in_sizes, int n_in,
                              void* d_out, int out_size, void* d_ws, size_t ws_size,
                              hipStream_t stream) {
  (void)in_sizes; (void)n_in; (void)out_size; (void)ws_size;
  const float* logits = (const float*)d_in[0];
  const int*   labels = (const int*)d_in[1];
  double* acc = (double*)d_ws;

  hiera_init<<<1, 32, 0, stream>>>(acc);
  hiera_main<<<MAIN_BLOCKS, 256, 0, stream>>>(logits, labels, acc);
  hiera_final<<<1, 32, 0, stream>>>(acc, (float*)d_out);
}